// ImprovedCombinedLoss_90245852823757
// MI455X (gfx1250) — compile-verified
//
#include <hip/hip_runtime.h>
#include <math.h>

typedef __attribute__((ext_vector_type(2))) float v2f;
typedef __attribute__((ext_vector_type(8))) float v8f;
typedef int v4i __attribute__((vector_size(16)));

#define LS_EPS   0.1f
#define MARGIN   0.3f
#define F_ALPHA  0.25f

#define AS1 __attribute__((address_space(1)))
#define AS3 __attribute__((address_space(3)))

#if __has_builtin(__builtin_amdgcn_global_load_async_to_lds_b128)
#define HAS_ASYNC_LDS 1
#else
#define HAS_ASYNC_LDS 0
#endif

// async 16-byte global -> LDS copy (CDNA5 GLOBAL_LOAD_ASYNC_TO_LDS_B128, ASYNCcnt)
__device__ __forceinline__ void async_copy16(const float* g, float* l) {
#if HAS_ASYNC_LDS
  __builtin_amdgcn_global_load_async_to_lds_b128(
      (AS1 v4i*)(AS1 void*)g, (AS3 v4i*)(AS3 void*)l, 0, 0);
#else
  *(float4*)l = *(const float4*)g;
#endif
}
__device__ __forceinline__ void wait_async() {
#if HAS_ASYNC_LDS
#if __has_builtin(__builtin_amdgcn_s_wait_asynccnt)
  __builtin_amdgcn_s_wait_asynccnt(0);
#else
  asm volatile("s_wait_asynccnt 0x0" ::: "memory");
#endif
#endif
}

// ---------------- reduction helpers (wave32-aware) ----------------
__device__ __forceinline__ float warp_sum(float v) {
  for (int m = 16; m; m >>= 1) v += __shfl_xor(v, m, 32);
  return v;
}
__device__ __forceinline__ float warp_max(float v) {
  for (int m = 16; m; m >>= 1) v = fmaxf(v, __shfl_xor(v, m, 32));
  return v;
}
__device__ float block_sum(float v, float* sb) {
  int lane = threadIdx.x & 31, w = threadIdx.x >> 5, nw = blockDim.x >> 5;
  v = warp_sum(v);
  if (lane == 0) sb[w] = v;
  __syncthreads();
  if (threadIdx.x == 0) { float s = 0.f; for (int i = 0; i < nw; ++i) s += sb[i]; sb[0] = s; }
  __syncthreads();
  float r = sb[0];
  __syncthreads();
  return r;
}
__device__ float block_max(float v, float* sb) {
  int lane = threadIdx.x & 31, w = threadIdx.x >> 5, nw = blockDim.x >> 5;
  v = warp_max(v);
  if (lane == 0) sb[w] = v;
  __syncthreads();
  if (threadIdx.x == 0) { float s = sb[0]; for (int i = 1; i < nw; ++i) s = fmaxf(s, sb[i]); sb[0] = s; }
  __syncthreads();
  float r = sb[0];
  __syncthreads();
  return r;
}

// ---------------- kernel 0: zero accumulators ----------------
__global__ void init_kernel(float* acc) {
  if (threadIdx.x < 8) acc[threadIdx.x] = 0.f;
}

// ---------------- kernel 1: log-softmax -> CE + focal ----------------
// acc[0]=ce_sum  acc[1]=focal_sum
__global__ void logits_kernel(const float* __restrict__ logits,
                              const int* __restrict__ labels,
                              float* acc, int C) {
  __shared__ float sb[32];
  const int row = blockIdx.x;
  const float* x = logits + (size_t)row * C;

  float mx = -3.4e38f;
  for (int c = threadIdx.x; c < C; c += blockDim.x) mx = fmaxf(mx, x[c]);
  mx = block_max(mx, sb);

  float se = 0.f, sx = 0.f;
  for (int c = threadIdx.x; c < C; c += blockDim.x) {
    float v = x[c];
    se += expf(v - mx);
    sx += v;
  }
  se = block_sum(se, sb);
  sx = block_sum(sx, sb);

  if (threadIdx.x == 0) {
    float M = mx + logf(se);                 // log-sum-exp
    float lpy = x[labels[row]] - M;          // log p(y)
    float mean_lp = sx / (float)C - M;       // mean_c log p(c)
    float ce_row = -(1.f - LS_EPS) * lpy - LS_EPS * mean_lp;
    float pt = expf(lpy);
    float om = 1.f - pt;
    float focal_row = F_ALPHA * om * om * lpy;  // gamma = 2, term <= 0 (faithful)
    atomicAdd(&acc[0], ce_row);
    atomicAdd(&acc[1], focal_row);
  }
}

// ---------------- kernel 2: fused row reductions ----------------
// sq[i] = ||f_i||^2 ; acc[4]=cen_sum acc[5]=align_sum acc[6]=ent_sum acc[7]=sd_sum
__global__ void rows_kernel(const float* __restrict__ feat,
                            const float* __restrict__ rgb,
                            const float* __restrict__ irf,
                            const float* __restrict__ mw,
                            const float* __restrict__ centers,
                            const int* __restrict__ labels,
                            float* acc, float* sq, int D) {
  __shared__ float sb[32];
  const int i = blockIdx.x;
  const float* f = feat + (size_t)i * D;
  const float* r = rgb  + (size_t)i * D;
  const float* m = irf  + (size_t)i * D;
  const float* cen = centers + (size_t)labels[i] * D;

  float s_sq = 0.f, s_cen = 0.f, s_dot = 0.f, s_nr = 0.f, s_ni = 0.f, s_sd = 0.f;
  for (int d = threadIdx.x; d < D; d += blockDim.x) {
    float fv = f[d], rv = r[d], iv = m[d], cv = cen[d];
    s_sq  += fv * fv;
    float dc = fv - cv;  s_cen += dc * dc;
    s_dot += rv * iv;
    s_nr  += rv * rv;
    s_ni  += iv * iv;
    float dd = rv - iv;  s_sd += dd * dd;
  }
  s_sq  = block_sum(s_sq,  sb);
  s_cen = block_sum(s_cen, sb);
  s_dot = block_sum(s_dot, sb);
  s_nr  = block_sum(s_nr,  sb);
  s_ni  = block_sum(s_ni,  sb);
  s_sd  = block_sum(s_sd,  sb);

  if (threadIdx.x == 0) {
    sq[i] = s_sq;
    atomicAdd(&acc[4], s_cen);
    float denom = fmaxf(sqrtf(s_nr), 1e-12f) * fmaxf(sqrtf(s_ni), 1e-12f);
    atomicAdd(&acc[5], 1.f - s_dot / denom);
    atomicAdd(&acc[7], s_sd);
    float w0 = mw[2 * i], w1 = mw[2 * i + 1];
    float ent = -(w0 * logf(w0 + 1e-8f) + w1 * logf(w1 + 1e-8f));
    atomicAdd(&acc[6], ent);
  }
}

// ---------------- kernel 3: WMMA Gram + fused batch-hard triplet ----------------
// acc[2]=triplet_sum  acc[3]=valid_count
// KC=32, LSTR=36: even stride -> 8B-aligned b64 fragment reads; 36r mod 64 banks
// distinct for r=0..15 -> conflict-free; (36r+4c)*4 = 16-byte aligned b128 targets.
#define KC    32
#define LSTR  36
#define NB128 (16 * KC / 4)   // 128 b128 copies per tile

__global__ void __launch_bounds__(256)
triplet_kernel(const float* __restrict__ feat,
               const int* __restrict__ labels,
               const float* __restrict__ sq,
               float* acc, int D, int B) {
  __shared__ float As[2][16 * LSTR];       // double-buffered row-tile chunk
  __shared__ float Bs[2][8][16 * LSTR];    // double-buffered per-wave col-tile chunk
  __shared__ float hpW[8][16];
  __shared__ float hnW[8][16];
  __shared__ int   labM[16];

  const int tid  = threadIdx.x;
  const int wave = tid >> 5;
  const int lane = tid & 31;
  const int half = lane >> 4;              // 0: lanes 0-15, 1: lanes 16-31
  const int l16  = lane & 15;
  const int m0   = blockIdx.x * 16;

  if (tid < 16) labM[tid] = labels[m0 + tid];

  float hp[8], hn[8], sqM[8];
  int   lbM[8];
  const int ncol_tiles   = B / 16;
  const int cts_per_wave = ncol_tiles / 8;   // uniform across waves -> safe barriers

  #pragma unroll
  for (int v = 0; v < 8; ++v) { hp[v] = -1.0f; hn[v] = 1e9f; }
  #pragma unroll
  for (int v = 0; v < 8; ++v) sqM[v] = sq[m0 + v + 8 * half];

  __syncthreads();   // labM visible
  #pragma unroll
  for (int v = 0; v < 8; ++v) lbM[v] = labM[v + 8 * half];

  // issue one K-chunk's worth of async global->LDS copies into buffer `buf`
  auto load_tiles = [&](int buf, int kc, int n0) {
    // A tile: 16 x KC, one b128 per thread for tid<NB128 (waves 0-3, wave-uniform)
    if (tid < NB128) {
      int rr = tid / (KC / 4);
      int cc = (tid % (KC / 4)) * 4;
      async_copy16(&feat[(size_t)(m0 + rr) * D + kc + cc],
                   &As[buf][rr * LSTR + cc]);
    }
    // B tile: per wave, 4 b128 per lane (coalesced along cc then rr)
    for (int t = lane; t < NB128; t += 32) {
      int rr = t / (KC / 4);
      int cc = (t % (KC / 4)) * 4;
      async_copy16(&feat[(size_t)(n0 + rr) * D + kc + cc],
                   &Bs[buf][wave][rr * LSTR + cc]);
    }
  };

  for (int ct = 0; ct < cts_per_wave; ++ct) {
    const int n0 = (ct * 8 + wave) * 16;
    v8f accv = {};

    // prologue: stage chunk 0
    load_tiles(0, 0, n0);
    wait_async();
    __syncthreads();

    int cur = 0;
    for (int kc = 0; kc < D; kc += KC) {
      const int nxt = cur ^ 1;
      // prefetch next chunk asynchronously while computing on `cur`
      if (kc + KC < D) load_tiles(nxt, kc + KC, n0);

      // 8 WMMAs per chunk on the full-precision f32 matrix pipe
      #pragma unroll
      for (int k = 0; k < KC; k += 4) {
        v2f a2 = *(const v2f*)&As[cur][l16 * LSTR + k + 2 * half];
        v2f b2 = *(const v2f*)&Bs[cur][wave][l16 * LSTR + k + 2 * half];
        accv = __builtin_amdgcn_wmma_f32_16x16x4_f32(
            false, a2, false, b2, (short)0, accv, false, false);
      }

      wait_async();      // our async copies into `nxt` have landed
      __syncthreads();   // all waves done reading `cur` and done loading
      cur = nxt;
    }

    // epilogue: distances, masks, butterfly hard-pos/neg over the 16 columns
    const int   j    = n0 + l16;
    const int   labj = labels[j];
    const float sqj  = sq[j];
    #pragma unroll
    for (int v = 0; v < 8; ++v) {
      const int i = m0 + v + 8 * half;
      float d2 = sqM[v] + sqj - 2.0f * accv[v];
      d2 = fmaxf(d2, 0.0f);
      float dist = (d2 > 0.0f) ? sqrtf(d2) : 0.0f;
      bool pos = (lbM[v] == labj) && (i != j);
      bool neg = (lbM[v] != labj);
      float pv = pos ? dist : -1.0f;
      float nv = neg ? dist : 1e9f;
      // reduce across the 16 lanes of this half (xor masks keep bit4 fixed)
      #pragma unroll
      for (int msk = 1; msk < 16; msk <<= 1) {
        pv = fmaxf(pv, __shfl_xor(pv, msk, 32));
        nv = fminf(nv, __shfl_xor(nv, msk, 32));
      }
      hp[v] = fmaxf(hp[v], pv);
      hn[v] = fminf(hn[v], nv);
    }
  }

  // combine across the 8 waves (each covered disjoint column tiles)
  if (l16 == 0) {
    #pragma unroll
    for (int v = 0; v < 8; ++v) {
      hpW[wave][v + 8 * half] = hp[v];
      hnW[wave][v + 8 * half] = hn[v];
    }
  }
  __syncthreads();
  if (tid < 16) {
    float HP = -1.0f, HN = 1e9f;
    for (int w = 0; w < 8; ++w) {
      HP = fmaxf(HP, hpW[w][tid]);
      HN = fminf(HN, hnW[w][tid]);
    }
    bool valid = (HP >= 0.0f) && (HN < 1e8f);   // matches ref sentinel semantics
    if (valid) {
      float tl = fmaxf(HP - HN + MARGIN, 0.0f);
      atomicAdd(&acc[2], tl);
      atomicAdd(&acc[3], 1.0f);
    }
  }
}

// ---------------- kernel 4: finalize ----------------
__global__ void finalize_kernel(const float* acc, float* out, int B, int D) {
  if (threadIdx.x == 0) {
    const float Bf = (float)B, BD = (float)B * (float)D;
    float ce      = acc[0] / Bf;
    float focal   = acc[1] / Bf;
    float triplet = acc[2] / fmaxf(acc[3], 1.0f);
    float cen     = acc[4] / BD;
    float align   = acc[5] / Bf;
    float balance = logf(2.0f) - acc[6] / Bf;
    float cons    = 0.25f * acc[7] / BD;   // algebraic reduction of modality consistency
    float total = (0.7f * ce + 0.3f * focal) + triplet + 0.0005f * cen
                + 0.5f * align + 0.1f * balance + 0.1f * cons;
    out[0] = total; out[1] = ce; out[2] = focal; out[3] = triplet;
    out[4] = cen;   out[5] = align; out[6] = balance; out[7] = cons;
  }
}

// ---------------- launcher ----------------
extern "C" void kernel_launch(void* const* d_in, const int* in_sizes, int n_in,
                              void* d_out, int out_size, void* d_ws, size_t ws_size,
                              hipStream_t stream) {
  const float* logits   = (const float*)d_in[0];
  const float* features = (const float*)d_in[1];
  const float* feat_rgb = (const float*)d_in[2];
  const float* feat_ir  = (const float*)d_in[3];
  const float* mw       = (const float*)d_in[4];
  const float* centers  = (const float*)d_in[5];
  const int*   labels   = (const int*)d_in[6];

  const int B = in_sizes[6];          // 1024
  const int C = in_sizes[0] / B;      // 8000
  const int D = in_sizes[1] / B;      // 2048

  float* acc = (float*)d_ws;          // 8 accumulators
  float* sqv = acc + 16;              // ||f_i||^2, B floats

  init_kernel<<<1, 32, 0, stream>>>(acc);
  logits_kernel<<<B, 256, 0, stream>>>(logits, labels, acc, C);
  rows_kernel<<<B, 256, 0, stream>>>(features, feat_rgb, feat_ir, mw, centers,
                                     labels, acc, sqv, D);
  triplet_kernel<<<B / 16, 256, 0, stream>>>(features, labels, sqv, acc, D, B);
  finalize_kernel<<<1, 32, 0, stream>>>(acc, (float*)d_out, B, D);
}